// StringGNNTail_6923487282242
// MI455X (gfx1250) — compile-verified
//
#include <hip/hip_runtime.h>

#define D    256
#define LDST 258   // W^T stride: banks (2n+k)%64 -> conflict-free, even -> b64-aligned
#define ASTR 260   // A-tile stride: %4==0 -> b128-aligned fills; reads hit 4*l15+khalf banks

typedef float v2f __attribute__((ext_vector_type(2)));
typedef float v8f __attribute__((ext_vector_type(8)));
typedef int   v4i __attribute__((vector_size(16)));                 // matches builtin param
typedef __attribute__((address_space(1))) v4i GASv4;                // global int32x4
typedef __attribute__((address_space(3))) v4i LASv4;                // LDS    int32x4

#if defined(__AMDGCN__) && __has_builtin(__builtin_amdgcn_global_load_async_to_lds_b128)
#define USE_ASYNC_LDS 1
#endif

// ---------------------------------------------------------------- degree ----
__global__ void init_deg_kernel(float* __restrict__ deg, int N) {
    int i = blockIdx.x * blockDim.x + threadIdx.x;
    if (i < N) deg[i] = 1.0f;                       // self-loop weight
}

__global__ void deg_edges_kernel(const int* __restrict__ ei, const float* __restrict__ ew,
                                 float* __restrict__ deg, int E) {
    int e = blockIdx.x * blockDim.x + threadIdx.x;
    if (e < E) unsafeAtomicAdd(&deg[ei[e]], ew[e]); // ei row 0 = dst
}

__global__ void dinv_kernel(float* __restrict__ dinv, int N) {
    int i = blockIdx.x * blockDim.x + threadIdx.x;
    if (i < N) {
        float d = dinv[i];
        dinv[i] = (d > 0.0f) ? rsqrtf(d) : 0.0f;
    }
}

// --------------------------------------------- async A-tile staging ---------
__device__ __forceinline__ void fillA(float* __restrict__ dstLds, const float* __restrict__ X,
                                      int mt, int nrows, int tid) {
#pragma unroll
    for (int j = 0; j < 2; ++j) {
        int c  = tid + j * 512;            // 1024 float4 chunks = 16 rows x 256
        int m  = c >> 6;                   // row in tile
        int k4 = (c & 63) << 2;            // col (x4)
        int row = mt * 16 + m;
        if (row >= nrows) row = nrows - 1; // tail clamp
        const float* g = X + (size_t)row * D + k4;
        float*       l = dstLds + m * ASTR + k4;
#if defined(USE_ASYNC_LDS)
        __builtin_amdgcn_global_load_async_to_lds_b128((GASv4*)g, (LASv4*)l, 0, 0);
#else
        *(float4*)l = *(const float4*)g;
#endif
    }
}

__device__ __forceinline__ void wait_fillA() {
#if defined(USE_ASYNC_LDS)
#if __has_builtin(__builtin_amdgcn_s_wait_asynccnt)
    __builtin_amdgcn_s_wait_asynccnt(0);
#else
    asm volatile("s_wait_asynccnt 0" ::: "memory");
#endif
#endif
}

// ------------------------------------------------------- WMMA f32 GEMM ------
// Y[M,256] = act(X[M,256] @ Wm[256,256] + bias), 16 waves/block.
// W^T resident in LDS (B fragments get hoisted to VGPRs); A tiles double-
// buffered in LDS via async-to-LDS DMA.
template <bool RELU>
__global__ __launch_bounds__(512)
void gemm256_wmma(const float* __restrict__ X, const float* __restrict__ Wm,
                  const float* __restrict__ bias, float* __restrict__ Y,
                  int ntiles, int nrows) {
    extern __shared__ float lds[];
    float* ldsW  = lds;                       // 256*LDST : ldsW[n*LDST+k] = W[k][n]
    float* ldsA0 = lds + D * LDST;            // 16*ASTR
    float* ldsA1 = ldsA0 + 16 * ASTR;         // 16*ASTR

    const int tid = threadIdx.x;
    for (int idx = tid; idx < D * D; idx += 512) {
        int k = idx >> 8;
        int n = idx & (D - 1);
        ldsW[n * LDST + k] = Wm[idx];
    }
    fillA(ldsA0, X, blockIdx.x, nrows, tid);  // prologue fill, buffer 0
    wait_fillA();
    __syncthreads();

    const int lane  = tid & 31;
    const int wid   = tid >> 5;               // 0..15 -> N strip
    const int n0    = wid << 4;
    const int l15   = lane & 15;              // A: M row; B/C/D: N col
    const int half  = lane >> 4;              // 0/1
    const int khalf = half << 1;              // K sub-offset {0,2}

    const float* wrow = ldsW + (n0 + l15) * LDST + khalf;
    const float  bv   = bias ? bias[n0 + l15] : 0.0f;

    int cur = 0;
    for (int mt = blockIdx.x; mt < ntiles; mt += gridDim.x) {
        const float* abase = (cur ? ldsA1 : ldsA0);
        float*       nbuf  = (cur ? ldsA0 : ldsA1);
        const float* arow  = abase + l15 * ASTR + khalf;

        int mtn = mt + gridDim.x;             // block-uniform
        if (mtn < ntiles) fillA(nbuf, X, mtn, nrows, tid);   // overlap next tile

        v8f c = {};
#pragma unroll
        for (int kk = 0; kk < D / 4; ++kk) {  // 64 x v_wmma_f32_16x16x4_f32
            v2f a = *(const v2f*)(arow + kk * 4);
            v2f b = *(const v2f*)(wrow + kk * 4);
            c = __builtin_amdgcn_wmma_f32_16x16x4_f32(
                    false, a, false, b, (short)0, c, false, false);
        }

        const int mbase = mt * 16 + half * 8; // D layout: VGPR v -> M = half*8 + v
#pragma unroll
        for (int v = 0; v < 8; ++v) {
            int mr = mbase + v;
            if (mr < nrows) {
                float r = c[v] + bv;
                if (RELU) r = fmaxf(r, 0.0f);
                Y[(size_t)mr * D + n0 + l15] = r;
            }
        }

        wait_fillA();
        __syncthreads();
        cur ^= 1;
    }
}

// ------------------------------------------------- self-loop + bias init ----
__global__ void selfloop_bias_kernel(const float* __restrict__ xw, const float* __restrict__ dinv,
                                     const float* __restrict__ b, float* __restrict__ h7, int N) {
    int idx = blockIdx.x * blockDim.x + threadIdx.x;    // one float4 per thread
    int total = N * (D / 4);
    if (idx >= total) return;
    int i  = idx >> 6;
    int j4 = (idx & 63) << 2;
    float s = dinv[i];
    s = s * s;                                          // self-loop norm (weight 1)
    float4 x  = *(const float4*)(xw + (size_t)i * D + j4);
    float4 bb = *(const float4*)(b + j4);
    float4 r;
    r.x = fmaf(s, x.x, bb.x);
    r.y = fmaf(s, x.y, bb.y);
    r.z = fmaf(s, x.z, bb.z);
    r.w = fmaf(s, x.w, bb.w);
    *(float4*)(h7 + (size_t)i * D + j4) = r;
}

// -------------------------------------------------------- edge scatter ------
// one wave per edge; each lane handles 8 contiguous features
__global__ __launch_bounds__(256)
void scatter_edges_kernel(const int* __restrict__ ei, const float* __restrict__ ew,
                          const float* __restrict__ dinv, const float* __restrict__ xw,
                          float* __restrict__ h7, int E) {
    int e = blockIdx.x * (blockDim.x >> 5) + (threadIdx.x >> 5);
    if (e >= E) return;
    int lane = threadIdx.x & 31;
    int dst = ei[e];                                   // row
    int src = ei[E + e];                               // col
    float norm = dinv[dst] * ew[e] * dinv[src];
    const float* xs = xw + (size_t)src * D + lane * 8;
    float*       hd = h7 + (size_t)dst * D + lane * 8;
    float4 x0 = *(const float4*)(xs);
    float4 x1 = *(const float4*)(xs + 4);
    unsafeAtomicAdd(hd + 0, norm * x0.x);
    unsafeAtomicAdd(hd + 1, norm * x0.y);
    unsafeAtomicAdd(hd + 2, norm * x0.z);
    unsafeAtomicAdd(hd + 3, norm * x0.w);
    unsafeAtomicAdd(hd + 4, norm * x1.x);
    unsafeAtomicAdd(hd + 5, norm * x1.y);
    unsafeAtomicAdd(hd + 6, norm * x1.z);
    unsafeAtomicAdd(hd + 7, norm * x1.w);
}

// ---------------------------------------------------------------- launch ----
extern "C" void kernel_launch(void* const* d_in, const int* in_sizes, int n_in,
                              void* d_out, int out_size, void* d_ws, size_t ws_size,
                              hipStream_t stream) {
    const float* h6 = (const float*)d_in[0];
    const int*   ei = (const int*)d_in[1];
    const float* ew = (const float*)d_in[2];
    const float* W  = (const float*)d_in[3];
    const float* b  = (const float*)d_in[4];
    const float* W1 = (const float*)d_in[5];
    const float* b1 = (const float*)d_in[6];
    const float* W2 = (const float*)d_in[7];
    const float* b2 = (const float*)d_in[8];

    const int N = in_sizes[0] / D;
    const int E = in_sizes[2];

    float* wsf  = (float*)d_ws;
    float* dinv = wsf;                         // N floats (deg -> dinv in place)
    float* xw   = wsf + N;                     // N*D floats (later reused as relu(h))
    float* h7   = xw + (size_t)N * D;          // N*D floats

    const int ntiles = (N + 15) / 16;
    const size_t ldsBytes = (size_t)(D * LDST + 2 * 16 * ASTR) * sizeof(float); // ~290 KB
    (void)hipFuncSetAttribute(reinterpret_cast<const void*>(&gemm256_wmma<false>),
                              hipFuncAttributeMaxDynamicSharedMemorySize, (int)ldsBytes);
    (void)hipFuncSetAttribute(reinterpret_cast<const void*>(&gemm256_wmma<true>),
                              hipFuncAttributeMaxDynamicSharedMemorySize, (int)ldsBytes);

    // 1) degree with self-loops, 2) dinv = rsqrt(deg)
    init_deg_kernel<<<(N + 255) / 256, 256, 0, stream>>>(dinv, N);
    deg_edges_kernel<<<(E + 255) / 256, 256, 0, stream>>>(ei, ew, dinv, E);
    dinv_kernel<<<(N + 255) / 256, 256, 0, stream>>>(dinv, N);

    // 3) xw = h6 @ W  (no bias)
    int gblocks = ntiles < 1024 ? ntiles : 1024;
    gemm256_wmma<false><<<gblocks, 512, ldsBytes, stream>>>(h6, W, nullptr, xw, ntiles, N);

    // 4) h7 = dinv^2 * xw + b   (self loop + bias)
    selfloop_bias_kernel<<<(N * (D / 4) + 255) / 256, 256, 0, stream>>>(xw, dinv, b, h7, N);

    // 5) h7 += scatter-add over edges
    scatter_edges_kernel<<<(E + 7) / 8, 256, 0, stream>>>(ei, ew, dinv, xw, h7, E);

    // 6) h = relu(h7 @ W1 + b1)  (reuse xw buffer)
    gemm256_wmma<true><<<gblocks, 512, ldsBytes, stream>>>(h7, W1, b1, xw, ntiles, N);

    // 7) out = h @ W2 + b2
    gemm256_wmma<false><<<gblocks, 512, ldsBytes, stream>>>(xw, W2, b2, (float*)d_out, ntiles, N);
}